// PosHead_43800076485371
// MI455X (gfx1250) — compile-verified
//
#include <hip/hip_runtime.h>

// ---------------- problem constants ----------------
#define B_IMG   64
#define H_IMG   640
#define W_IMG   640
#define K_TOP   6
#define RAD     4          // (9x9 window)/2
#define CONF    0.3f

// ---------------- tiling ----------------
#define ROWS       16                   // output rows per strip
#define STRIPS     (H_IMG / ROWS)       // 40 strips per image
#define HALO_ROWS  (ROWS + 2 * RAD)     // 24 rows staged in LDS
#define CAND_CAP   512
#define T1         256
#define T2         256

typedef __attribute__((ext_vector_type(4))) unsigned int u32x4;
typedef __attribute__((ext_vector_type(8))) int          i32x8;
typedef __attribute__((ext_vector_type(4))) int          i32x4;

// =====================================================================
// Kernel 1: per (image, strip): TDM-stage strip into LDS, separable 9x9
// max, peak test, per-strip top-6 -> workspace.
// =====================================================================
__global__ void __launch_bounds__(T1) pos_head_stage1(
    const float* __restrict__ hm,
    float* __restrict__ wsScore,
    int*   __restrict__ wsIdx)
{
    extern __shared__ float smem[];
    float* raw   = smem;                                  // HALO_ROWS * W
    float* hmax  = smem + HALO_ROWS * W_IMG;              // HALO_ROWS * W
    float* candS = hmax + HALO_ROWS * W_IMG;              // CAND_CAP
    int*   candI = (int*)(candS + CAND_CAP);              // CAND_CAP

    __shared__ float redS[T1];
    __shared__ int   redP[T1];
    __shared__ int   candCount;

    const int tid    = threadIdx.x;
    const int img    = blockIdx.x / STRIPS;
    const int strip  = blockIdx.x % STRIPS;
    const int row0   = strip * ROWS;
    const int loadLo = (row0 - RAD) > 0 ? (row0 - RAD) : 0;
    const int loadHiT = row0 + ROWS + RAD;
    const int loadHi = loadHiT < H_IMG ? loadHiT : H_IMG;
    const int nLoad  = loadHi - loadLo;

    if (tid == 0) candCount = 0;

    const unsigned long long gbase =
        ((unsigned long long)img * H_IMG + (unsigned long long)loadLo) * W_IMG;

#if defined(__gfx1250__) && __has_builtin(__builtin_amdgcn_tensor_load_to_lds)
    // ---- Tensor Data Mover: DMA the strip (nLoad x 640 f32) into LDS ----
    if (tid < 32) {  // one wave issues the TDM op (1 issue per wave)
        unsigned long long ga = (unsigned long long)(uintptr_t)(hm + gbase);
        unsigned int lds_addr = (unsigned int)(uintptr_t)raw;  // LDS byte offset

        u32x4 g0;
        g0.x = 1u;                                   // count=1, user mode
        g0.y = lds_addr;                             // lds_addr [63:32]
        g0.z = (unsigned int)ga;                     // global_addr [95:64]
        g0.w = (unsigned int)((ga >> 32) & 0x01FFFFFFu) | (2u << 30); // [120:96], type=2

        i32x8 g1;
        g1[0] = (2 << 16);                                   // data_size = 4B
        g1[1] = (int)((W_IMG & 0xFFFF) << 16);               // tensor_dim0 lo16 @ [63:48]
        g1[2] = (int)(((W_IMG >> 16) & 0xFFFF)               // tensor_dim0 hi16
                      | ((nLoad & 0xFFFF) << 16));           // tensor_dim1 lo16
        g1[3] = (int)(((nLoad >> 16) & 0xFFFF)               // tensor_dim1 hi16
                      | ((W_IMG & 0xFFFF) << 16));           // tile_dim0 @ [127:112]
        g1[4] = (int)(nLoad & 0xFFFF);                       // tile_dim1; tile_dim2=0
        g1[5] = W_IMG;                                       // tensor_dim0_stride lo32
        g1[6] = (int)(((W_IMG * H_IMG) & 0xFFFF) << 16);     // dim1_stride lo16 @ [223:208]
        g1[7] = (int)((W_IMG * H_IMG) >> 16);                // dim1_stride [47:16]

        i32x4 gz4 = {0, 0, 0, 0};
        i32x8 gz8 = {0, 0, 0, 0, 0, 0, 0, 0};
        __builtin_amdgcn_tensor_load_to_lds(g0, g1, gz4, gz4, gz8, 0);
#if __has_builtin(__builtin_amdgcn_s_wait_tensorcnt)
        __builtin_amdgcn_s_wait_tensorcnt(0);
#else
        asm volatile("s_wait_tensorcnt 0x0" ::: "memory");
#endif
    }
#else
    for (int p = tid; p < nLoad * W_IMG; p += T1)
        raw[p] = hm[gbase + p];
#endif
    __syncthreads();

    // ---- Stage A: horizontal 9-max (column-clamped) ----
    const int nPix = nLoad * W_IMG;
    for (int p = tid; p < nPix; p += T1) {
        int c  = p % W_IMG;
        int lo = (c - RAD > 0 ? c - RAD : 0) - c;
        int hi = (c + RAD < W_IMG - 1 ? c + RAD : W_IMG - 1) - c;
        float m = raw[p + lo];
        for (int d = lo + 1; d <= hi; ++d) m = fmaxf(m, raw[p + d]);
        hmax[p] = m;
    }
    __syncthreads();

    // ---- Stage B: vertical 9-max + peak test, collect candidates ----
    for (int p = tid; p < ROWS * W_IMG; p += T1) {
        int lrOut = p / W_IMG;
        int c     = p % W_IMG;
        int rG    = row0 + lrOut;
        int lr    = rG - loadLo;
        float center = raw[lr * W_IMG + c];
        int rlo = lr - RAD > 0 ? lr - RAD : 0;
        int rhi = lr + RAD < nLoad - 1 ? lr + RAD : nLoad - 1;
        float m = hmax[rlo * W_IMG + c];
        for (int rr = rlo + 1; rr <= rhi; ++rr)
            m = fmaxf(m, hmax[rr * W_IMG + c]);
        if (center > CONF && center == m) {
            int pos = atomicAdd(&candCount, 1);
            if (pos < CAND_CAP) {
                candS[pos] = center;
                candI[pos] = rG * W_IMG + c;
            }
        }
    }
    __syncthreads();

    // ---- Stage C: per-strip top-6 via 6 argmax reductions ----
    int n = candCount < CAND_CAP ? candCount : CAND_CAP;
    float* outS = wsScore + ((size_t)img * STRIPS + strip) * K_TOP;
    int*   outI = wsIdx   + ((size_t)img * STRIPS + strip) * K_TOP;

    for (int k = 0; k < K_TOP; ++k) {
        float ls = -1.0f; int lp = -1;
        for (int j = tid; j < n; j += T1) {
            float s = candS[j];
            if (s > ls) { ls = s; lp = j; }
        }
        redS[tid] = ls; redP[tid] = lp;
        __syncthreads();
        for (int off = T1 / 2; off > 0; off >>= 1) {
            if (tid < off) {
                if (redS[tid + off] > redS[tid]) {
                    redS[tid] = redS[tid + off];
                    redP[tid] = redP[tid + off];
                }
            }
            __syncthreads();
        }
        if (tid == 0) {
            float s = redS[0]; int p = redP[0];
            if (p >= 0 && s > CONF) {
                outS[k] = s; outI[k] = candI[p];
                candS[p] = -2.0f;  // mark consumed
            } else {
                outS[k] = 0.0f; outI[k] = 0;
            }
        }
        __syncthreads();
    }
}

// =====================================================================
// Kernel 2: per image, merge STRIPS*K candidates -> final top-6, map to
// (x,y) in [0,1] via linspace, apply validity, write outputs.
// =====================================================================
__global__ void __launch_bounds__(T2) pos_head_stage2(
    const float* __restrict__ wsScore,
    const int*   __restrict__ wsIdx,
    float* __restrict__ out)
{
    __shared__ float cs[T2];
    __shared__ int   ci[T2];
    __shared__ float rs[T2];
    __shared__ int   rp[T2];

    const int tid = threadIdx.x;
    const int img = blockIdx.x;
    const int nc  = STRIPS * K_TOP;  // 240 candidates per image

    if (tid < nc) {
        cs[tid] = wsScore[(size_t)img * nc + tid];
        ci[tid] = wsIdx[(size_t)img * nc + tid];
    } else {
        cs[tid] = -2.0f;
        ci[tid] = 0;
    }
    __syncthreads();

    float* outPos  = out;                          // B*K*2 floats
    float* outConf = out + (size_t)B_IMG * K_TOP * 2;

    for (int k = 0; k < K_TOP; ++k) {
        rs[tid] = cs[tid]; rp[tid] = tid;
        __syncthreads();
        for (int off = T2 / 2; off > 0; off >>= 1) {
            if (tid < off) {
                if (rs[tid + off] > rs[tid]) {
                    rs[tid] = rs[tid + off];
                    rp[tid] = rp[tid + off];
                }
            }
            __syncthreads();
        }
        if (tid == 0) {
            float s = rs[0]; int p = rp[0];
            bool valid = s > CONF;
            int idx = valid ? ci[p] : 0;
            float px = valid ? (float)(idx % W_IMG) * (1.0f / (W_IMG - 1)) : -1.0f;
            float py = valid ? (float)(idx / W_IMG) * (1.0f / (H_IMG - 1)) : -1.0f;
            outPos[((size_t)img * K_TOP + k) * 2 + 0] = px;
            outPos[((size_t)img * K_TOP + k) * 2 + 1] = py;
            outConf[(size_t)img * K_TOP + k] = valid ? s : 0.0f;
            cs[p] = -2.0f;  // mark consumed
        }
        __syncthreads();
    }
}

extern "C" void kernel_launch(void* const* d_in, const int* in_sizes, int n_in,
                              void* d_out, int out_size, void* d_ws, size_t ws_size,
                              hipStream_t stream)
{
    const float* hm = (const float*)d_in[0];

    float* wsScore = (float*)d_ws;
    int*   wsIdx   = (int*)((char*)d_ws +
                            (size_t)B_IMG * STRIPS * K_TOP * sizeof(float));

    size_t dynBytes = (size_t)HALO_ROWS * W_IMG * sizeof(float) * 2   // raw + hmax
                    + (size_t)CAND_CAP * (sizeof(float) + sizeof(int));

    pos_head_stage1<<<B_IMG * STRIPS, T1, dynBytes, stream>>>(hm, wsScore, wsIdx);
    pos_head_stage2<<<B_IMG, T2, 0, stream>>>(wsScore, wsIdx, (float*)d_out);
}